// SOM_27934467293842
// MI455X (gfx1250) — compile-verified
//
#include <hip/hip_runtime.h>
#include <stdint.h>

// Problem constants (match reference)
#define SOM_M     512
#define SOM_N     512
#define SOM_DIM   256
#define SOM_MN    (SOM_M * SOM_N)          // 262144
#define SOM_EPS   1e-6f
#define SOM_ALPHA 0.3f
#define SOM_SIGMA 256.0f

typedef __attribute__((ext_vector_type(2))) float v2f;
typedef __attribute__((ext_vector_type(8))) float v8f;

// ---------------------------------------------------------------------------
// Kernel 0: initialize the global argmin cell (u64 packed key) to +inf
// ---------------------------------------------------------------------------
__global__ void som_init_kernel(unsigned long long* best) {
    if (threadIdx.x == 0 && blockIdx.x == 0) {
        best[0] = 0xFFFFFFFFFFFFFFFFull;
    }
}

// ---------------------------------------------------------------------------
// Kernel 1: BMU search. One wave (32 threads) per 16-row tile of `weights`.
//   dist2[r] = sum_k ((x[k]-w[r][k])+eps)^2  computed as the diagonal of
//   D = A * A^T using V_WMMA_F32_16X16X4_F32 accumulation over K (64 steps).
//
// f32 16x16x4 WMMA lane layout (wave32):
//   A (16x4):  v0 = lanes0-15 K=0 / lanes16-31 K=2 ; v1 = K=1 / K=3, M = lane%16
//   B (4x16):  v0 = lanes0-15 K=0 / lanes16-31 K=2 ; v1 = K=1 / K=3, N = lane%16
// -> A and B register contents are identical for B = A^T over the same rows.
// Each lane loads exactly its float2 slice from LDS (no cross-lane selects):
//   lane L      : d[row][4i+0 .. 4i+1]
//   lane L+16   : d[row][4i+2 .. 4i+3]
// Bank check: lane L uses banks 4r..4r+1, lane L+16 uses 4r+2..4r+3 -> all 64
// banks covered, conflict-free ds_load_b64.
// ---------------------------------------------------------------------------
__global__ __launch_bounds__(32) void som_bmu_kernel(
        const float* __restrict__ x,
        const float* __restrict__ weights,
        unsigned long long* __restrict__ best) {
    __shared__ float lds_w[16 * 260];   // 16 rows, stride 260 dwords (bank-safe, 16B-aligned)
    __shared__ float lds_x[SOM_DIM];
    __shared__ float lds_c[32 * 8];

    const int lane = threadIdx.x;          // 0..31
    const int tile = blockIdx.x;           // 0..MN/16-1
    const float4* wt = (const float4*)(weights + (size_t)tile * 16 * SOM_DIM);

    // Stage x (64 float4) -- coalesced
    for (int t = lane; t < SOM_DIM / 4; t += 32) {
        ((float4*)lds_x)[t] = ((const float4*)x)[t];
    }
    // Stage 16x256 weight tile (1024 float4) -- coalesced global reads
    for (int t = lane; t < 16 * (SOM_DIM / 4); t += 32) {
        int r = t >> 6;                    // row 0..15
        int c = t & 63;                    // float4 col
        ((float4*)(lds_w + r * 260))[c] = wt[t];
    }
    __syncthreads();

    const int row = lane & 15;
    const int ksel = (lane >> 4) << 1;     // 0 for low half, 2 for high half

    // Per-lane LDS base offsets (in floats)
    const float* wrow = lds_w + row * 260 + ksel;
    const float* xrow = lds_x + ksel;

    v8f acc = {};
    #pragma unroll 8
    for (int i = 0; i < SOM_DIM / 4; ++i) {
        float2 wv = *(const float2*)(wrow + 4 * i);   // ds_load_b64, conflict-free
        float2 xv = *(const float2*)(xrow + 4 * i);   // ds_load_b64, half-wave broadcast
        v2f ab;
        ab.x = (xv.x - wv.x) + SOM_EPS;
        ab.y = (xv.y - wv.y) + SOM_EPS;
        // D = A*B + C  (A == B registers; Gram matrix, diagonal = sum of squares)
        acc = __builtin_amdgcn_wmma_f32_16x16x4_f32(
                /*neg_a=*/false, ab, /*neg_b=*/false, ab,
                /*c_mod=*/(short)0, acc, /*reuse_a=*/false, /*reuse_b=*/false);
    }

    // Spill accumulator so each lane<16 can fetch its diagonal element.
    #pragma unroll
    for (int j = 0; j < 8; ++j) lds_c[lane * 8 + j] = acc[j];
    __syncthreads();

    // Diagonal (r,r): r<8 -> VGPR r of lane r ; r>=8 -> VGPR r-8 of lane r+16
    unsigned long long key = 0xFFFFFFFFFFFFFFFFull;
    if (lane < 16) {
        int r = lane;
        float d2 = (r < 8) ? lds_c[r * 8 + r] : lds_c[(r + 16) * 8 + (r - 8)];
        unsigned int bits = __float_as_uint(d2);         // d2 >= 0 -> monotonic
        unsigned int idx  = (unsigned int)tile * 16u + (unsigned int)r;
        key = ((unsigned long long)bits << 32) | (unsigned long long)idx;
    }
    // Wave min-reduction (64-bit via two 32-bit shuffles); ties -> smaller index
    #pragma unroll
    for (int off = 16; off >= 1; off >>= 1) {
        unsigned int lo = (unsigned int)key;
        unsigned int hb = (unsigned int)(key >> 32);
        lo = __shfl_xor(lo, off, 32);
        hb = __shfl_xor(hb, off, 32);
        unsigned long long other = ((unsigned long long)hb << 32) | lo;
        key = (other < key) ? other : key;
    }
    if (lane == 0) atomicMin(best, key);
}

// ---------------------------------------------------------------------------
// Kernel 2: weight update (grid-stride float4 stream; ~512 MB traffic)
//   new_w = w + alpha_op * exp(-locdist / (2*sigma_op^2)) * (x - w)
// locations[k] == (k / 512, k % 512) by construction.
// ---------------------------------------------------------------------------
__global__ void som_update_kernel(
        const float* __restrict__ x,
        const float* __restrict__ weights,
        const int*   __restrict__ it_ptr,
        const unsigned long long* __restrict__ best,
        float* __restrict__ out) {
    const unsigned int bmu = (unsigned int)(best[0] & 0xFFFFFFFFull);
    const int bi = (int)(bmu >> 9);
    const int bj = (int)(bmu & 511u);

    const float lr       = 1.0f - (float)(*it_ptr) / 100.0f;
    const float alpha_op = SOM_ALPHA * lr;
    const float sigma_op = SOM_SIGMA * lr;
    const float ndenom   = -1.0f / (2.0f * sigma_op * sigma_op);

    const float4* w4 = (const float4*)weights;
    const float4* x4 = (const float4*)x;
    float4*       o4 = (float4*)out;

    const unsigned int total4 = (unsigned int)SOM_MN * (SOM_DIM / 4);  // 16M
    const unsigned int stride = gridDim.x * blockDim.x;
    for (unsigned int g = blockIdx.x * blockDim.x + threadIdx.x; g < total4; g += stride) {
        unsigned int neuron = g >> 6;                  // / (DIM/4)
        int i = (int)(neuron >> 9);
        int j = (int)(neuron & 511u);
        int di = i - bi, dj = j - bj;
        float locdist = (float)(di * di + dj * dj);
        float lm = alpha_op * expf(ndenom * locdist);

        float4 wv = w4[g];
        float4 xv = x4[g & 63u];
        float4 ov;
        ov.x = fmaf(lm, xv.x - wv.x, wv.x);
        ov.y = fmaf(lm, xv.y - wv.y, wv.y);
        ov.z = fmaf(lm, xv.z - wv.z, wv.z);
        ov.w = fmaf(lm, xv.w - wv.w, wv.w);
        o4[g] = ov;
    }
}

// ---------------------------------------------------------------------------
extern "C" void kernel_launch(void* const* d_in, const int* in_sizes, int n_in,
                              void* d_out, int out_size, void* d_ws, size_t ws_size,
                              hipStream_t stream) {
    const float* x       = (const float*)d_in[0];   // (256,)
    const float* weights = (const float*)d_in[1];   // (262144, 256)
    // d_in[2] = locations (unused: derived analytically from index)
    const int*   it_ptr  = (const int*)d_in[3];     // scalar
    float*       out     = (float*)d_out;           // (262144, 256)

    unsigned long long* best = (unsigned long long*)d_ws;

    som_init_kernel<<<1, 32, 0, stream>>>(best);

    som_bmu_kernel<<<SOM_MN / 16, 32, 0, stream>>>(x, weights, best);

    som_update_kernel<<<4096, 256, 0, stream>>>(x, weights, it_ptr, best, out);
}